// PINN2_61864708932295
// MI455X (gfx1250) — compile-verified
//
#include <hip/hip_runtime.h>
#include <cmath>

// ---------------------------------------------------------------------------
// PINN2 (Burgers-style IRK PINN) forward pass for MI455X (gfx1250, wave32).
//
// All matmuls mapped to V_WMMA_F32_16X16X4_F32 with orientation
//   [M = features, N = samples(16/wave), K = in-features]
// so layer->layer fragment handoff is just half-wave ds_swizzle SWAPX16
// + lane-half selects (no LDS transpose between layers).
// ---------------------------------------------------------------------------

#define WIDTH 50      // hidden width
#define QQ    100     // output dim Q
#define HT    4       // hidden feature tiles (50 -> 64 padded)
#define QT    7       // output feature tiles (100 -> 112 padded)
#define KH    13      // K-steps over 50 in-features (padded to 52)
#define KF    25      // K-steps over 100 f-features (exact)
#define WPB   4       // waves per block
#define LSTR  113     // LDS row stride (floats), odd-ish to avoid bank conflicts

typedef float v2f __attribute__((ext_vector_type(2)));
typedef float v8f __attribute__((ext_vector_type(8)));

static __device__ __forceinline__ int imin(int a, int b) { return a < b ? a : b; }

// exchange value between lane halves (lane ^ 16) via DS_SWIZZLE_B32 SWAPX16
static __device__ __forceinline__ float swap16(float x) {
  int i = __builtin_bit_cast(int, x);
  i = __builtin_amdgcn_ds_swizzle(i, 0x401F);  // group-of-32: xor=0x10, and=0x1f
  return __builtin_bit_cast(float, i);
}

// D = A(16x4) * B(4x16) + C  in fp32
static __device__ __forceinline__ v8f wmma4(v2f a, v2f b, v8f c) {
  return __builtin_amdgcn_wmma_f32_16x16x4_f32(false, a, false, b, (short)0, c,
                                               false, false);
}

// Build a B-fragment (rows lm..lm+3 of a 16-row D-frag tile s).
// D layout: VGPR r = (M=r, N=lane) lanes 0-15 ; (M=r+8, N=lane-16) lanes 16-31.
// B layout: VGPR v = (K=v, N=lane) lanes 0-15 ; (K=v+2, N=lane-16) lanes 16-31.
static __device__ __forceinline__ v2f make_b(const v8f s, int lm, bool lh) {
  float p0, p1, q0, q1;
  if (lm == 0)      { p0 = s[0]; p1 = s[1]; q0 = s[2]; q1 = s[3]; }
  else if (lm == 4) { p0 = s[4]; p1 = s[5]; q0 = s[6]; q1 = s[7]; }
  else if (lm == 8) { q0 = s[0]; q1 = s[1]; p0 = s[2]; p1 = s[3]; }
  else              { q0 = s[4]; q1 = s[5]; p0 = s[6]; p1 = s[7]; }
  // evaluate swizzles unconditionally (convergent), then lane-half select
  float sq0 = swap16(q0), sq1 = swap16(q1);
  v2f b;
  if (lm == 0 || lm == 4) { b.x = lh ? sq0 : p0; b.y = lh ? sq1 : p1; }
  else                    { b.x = lh ? p0 : sq0; b.y = lh ? p1 : sq1; }
  return b;
}

// A-fragment = 16x4 weight tile, lane m holds row gm, two consecutive K values.
// Clamped loads: padded rows/cols multiply zeroed activation rows, so any
// in-bounds value is safe.
static __device__ __forceinline__ v2f afrag_w(const float* __restrict__ W,
                                              int rows, int cols, int gm, int gk) {
  gm = imin(gm, rows - 1);
  gk = imin(gk, cols - 2);
  const float* p = W + gm * cols + gk;  // gm*cols+gk is even -> 8B aligned
  v2f a;
  a.x = p[0];
  a.y = p[1];
  return a;
}

// Output-layer tile group: computes u,ux,uxx for tiles [T0, T0+NT), folds into
// u[] and f[] fragments. k-outer so B-frags built once per K-step.
template <int T0, int NT>
static __device__ __forceinline__ void out_group(
    const float* __restrict__ Wout, const float* __restrict__ bout,
    const v8f* h, const v8f* hx, const v8f* hxx, v8f* u, v8f* f,
    int m, bool lh, float l1v, float el2) {
  v8f au[NT], ax[NT], axx[NT];
#pragma unroll
  for (int i = 0; i < NT; i++) {
#pragma unroll
    for (int r = 0; r < 8; r++) {
      au[i][r]  = bout[imin(16 * (T0 + i) + r + (lh ? 8 : 0), QQ - 1)];
      ax[i][r]  = 0.f;
      axx[i][r] = 0.f;
    }
  }
#pragma unroll
  for (int k = 0; k < KH; k++) {
    const int lm = (k & 3) * 4, st = k >> 2;
    v2f bh  = make_b(h[st], lm, lh);
    v2f bx  = make_b(hx[st], lm, lh);
    v2f bxx = make_b(hxx[st], lm, lh);
#pragma unroll
    for (int i = 0; i < NT; i++) {
      v2f a = afrag_w(Wout, QQ, WIDTH, 16 * (T0 + i) + m, 4 * k + (lh ? 2 : 0));
      au[i]  = wmma4(a, bh, au[i]);
      ax[i]  = wmma4(a, bx, ax[i]);
      axx[i] = wmma4(a, bxx, axx[i]);
    }
  }
#pragma unroll
  for (int i = 0; i < NT; i++) {
    u[T0 + i] = au[i];
#pragma unroll
    for (int r = 0; r < 8; r++) {
      // f = -lam1 * u * u_x + exp(lam2) * u_xx
      f[T0 + i][r] = fmaf(-l1v * au[i][r], ax[i][r], el2 * axx[i][r]);
    }
  }
}

__global__ __launch_bounds__(128) void PINN2_61864708932295_kernel(
    const float* __restrict__ x, const float* __restrict__ dt,
    const float* __restrict__ alpha, const float* __restrict__ beta,
    const float* __restrict__ lam1, const float* __restrict__ lam2,
    const float* __restrict__ W1, const float* __restrict__ b1,
    const float* __restrict__ W2, const float* __restrict__ b2,
    const float* __restrict__ W3, const float* __restrict__ b3,
    const float* __restrict__ W4, const float* __restrict__ b4,
    const float* __restrict__ W5, const float* __restrict__ b5,
    const float* __restrict__ Wout, const float* __restrict__ bout,
    float* __restrict__ out, int Ntot) {
  __shared__ float lds[WPB * 16 * LSTR];

  const int wid  = threadIdx.x >> 5;
  const int lane = threadIdx.x & 31;
  const bool lh  = lane >= 16;
  const int m    = lane & 15;
  const int base = (blockIdx.x * WPB + wid) * 16;  // first sample of this wave
  if (base >= Ntot) return;                        // wave-uniform exit
  float* wlds = lds + wid * (16 * LSTR);

  const float dtv = dt[0];
  const float l1v = lam1[0];
  const float el2 = expf(lam2[0]);

  const float xv = x[imin(base + m, Ntot - 1)];

  // ---- layer 1 (in_dim = 1, pure elementwise in fragment layout) ----------
  v8f h[HT], hx[HT], hxx[HT];
#pragma unroll
  for (int t = 0; t < HT; t++) {
#pragma unroll
    for (int r = 0; r < 8; r++) {
      const int feat = 16 * t + r + (lh ? 8 : 0);
      const int fc   = imin(feat, WIDTH - 1);
      const float w  = W1[fc];
      const float z  = fmaf(w, xv, b1[fc]);
      const float a  = tanhf(z);
      const float s  = 1.f - a * a;
      const bool ok  = feat < WIDTH;  // zero padded rows -> safe clamped A loads
      h[t][r]   = ok ? a : 0.f;
      hx[t][r]  = ok ? s * w : 0.f;
      hxx[t][r] = ok ? (-2.f * a * s * w * w) : 0.f;
    }
  }

  // ---- hidden layers 2..5: three 50x50 GEMM streams each -------------------
  for (int l = 0; l < 4; l++) {
    const float* W = (l == 0) ? W2 : (l == 1) ? W3 : (l == 2) ? W4 : W5;
    const float* B = (l == 0) ? b2 : (l == 1) ? b3 : (l == 2) ? b4 : b5;
    v8f nz[HT], nzx[HT], nzxx[HT];
#pragma unroll
    for (int t = 0; t < HT; t++) {
#pragma unroll
      for (int r = 0; r < 8; r++) {
        nz[t][r]   = B[imin(16 * t + r + (lh ? 8 : 0), WIDTH - 1)];
        nzx[t][r]  = 0.f;
        nzxx[t][r] = 0.f;
      }
    }
#pragma unroll
    for (int k = 0; k < KH; k++) {
      const int lm = (k & 3) * 4, st = k >> 2;
      v2f bh  = make_b(h[st], lm, lh);
      v2f bx  = make_b(hx[st], lm, lh);
      v2f bxx = make_b(hxx[st], lm, lh);
#pragma unroll
      for (int t = 0; t < HT; t++) {
        v2f a = afrag_w(W, WIDTH, WIDTH, 16 * t + m, 4 * k + (lh ? 2 : 0));
        nz[t]   = wmma4(a, bh, nz[t]);
        nzx[t]  = wmma4(a, bx, nzx[t]);
        nzxx[t] = wmma4(a, bxx, nzxx[t]);
      }
    }
#pragma unroll
    for (int t = 0; t < HT; t++) {
#pragma unroll
      for (int r = 0; r < 8; r++) {
        const float a   = tanhf(nz[t][r]);
        const float s   = 1.f - a * a;
        const float zx  = nzx[t][r];
        const float zxx = nzxx[t][r];
        const int feat  = 16 * t + r + (lh ? 8 : 0);
        const bool ok   = feat < WIDTH;
        h[t][r]   = ok ? a : 0.f;
        hx[t][r]  = ok ? s * zx : 0.f;
        hxx[t][r] = ok ? (s * zxx - 2.f * a * s * zx * zx) : 0.f;
      }
    }
  }

  // ---- output layer (50 -> 100) + PDE residual f --------------------------
  v8f u[QT], f[QT];
  out_group<0, 4>(Wout, bout, h, hx, hxx, u, f, m, lh, l1v, el2);
  out_group<4, 3>(Wout, bout, h, hx, hxx, u, f, m, lh, l1v, el2);

  // ---- final GEMM: G[q,i] = sum_k (beta[k]-alpha[q,k]) * f[k,i] -----------
  v8f g[QT];
#pragma unroll
  for (int t = 0; t < QT; t++) {
#pragma unroll
    for (int r = 0; r < 8; r++) g[t][r] = 0.f;
  }
#pragma unroll
  for (int k = 0; k < KF; k++) {
    const int lm = (k & 3) * 4, st = k >> 2;
    const int gk = 4 * k + (lh ? 2 : 0);
    v2f bf = make_b(f[st], lm, lh);
    const float bk0 = beta[gk], bk1 = beta[gk + 1];
#pragma unroll
    for (int t = 0; t < QT; t++) {
      const int gq = imin(16 * t + m, QQ - 1);
      v2f a;
      a.x = bk0 - alpha[gq * QQ + gk];
      a.y = bk1 - alpha[gq * QQ + gk + 1];
      g[t] = wmma4(a, bf, g[t]);
    }
  }

  // ---- out = u + dt * G : stage per-wave in LDS, then coalesced store -----
#pragma unroll
  for (int t = 0; t < QT; t++) {
#pragma unroll
    for (int r = 0; r < 8; r++) {
      const int q = 16 * t + r + (lh ? 8 : 0);
      if (q < 112)  // always true; keeps indexing obvious
        wlds[m * LSTR + q] = u[t][r] + dtv * g[t][r];
    }
  }
  // same-wave LDS RAW: compiler inserts s_wait_dscnt
  for (int idx = lane; idx < 16 * QQ; idx += 32) {
    const int i = idx / QQ;
    const int q = idx - i * QQ;
    const int smp = base + i;
    if (smp < Ntot) out[smp * QQ + q] = wlds[i * LSTR + q];
  }
}

extern "C" void kernel_launch(void* const* d_in, const int* in_sizes, int n_in,
                              void* d_out, int out_size, void* d_ws, size_t ws_size,
                              hipStream_t stream) {
  const float* x     = (const float*)d_in[0];
  const float* dt    = (const float*)d_in[1];
  const float* alpha = (const float*)d_in[2];
  const float* beta  = (const float*)d_in[3];
  const float* lam1  = (const float*)d_in[4];
  const float* lam2  = (const float*)d_in[5];
  const float* W1 = (const float*)d_in[6],  *b1 = (const float*)d_in[7];
  const float* W2 = (const float*)d_in[8],  *b2 = (const float*)d_in[9];
  const float* W3 = (const float*)d_in[10], *b3 = (const float*)d_in[11];
  const float* W4 = (const float*)d_in[12], *b4 = (const float*)d_in[13];
  const float* W5 = (const float*)d_in[14], *b5 = (const float*)d_in[15];
  const float* Wout = (const float*)d_in[16], *bout = (const float*)d_in[17];

  const int Ntot   = in_sizes[0];           // x is [N,1]
  const int tiles  = (Ntot + 15) / 16;      // 16 samples per wave
  const int blocks = (tiles + WPB - 1) / WPB;

  PINN2_61864708932295_kernel<<<blocks, WPB * 32, 0, stream>>>(
      x, dt, alpha, beta, lam1, lam2, W1, b1, W2, b2, W3, b3, W4, b4, W5, b5,
      Wout, bout, (float*)d_out, Ntot);
}